// WindowAttention_77713138254290
// MI455X (gfx1250) — compile-verified
//
#include <hip/hip_runtime.h>
#include <math.h>

// ---------------------------------------------------------------------------
// Fused shifted-window attention for MI455X (gfx1250, wave32, WMMA).
// One workgroup = one (batch, window). Everything between the input roll and
// the output roll stays in LDS; HBM traffic = read x once + write out once.
// All GEMMs use V_WMMA_F32_16X16X4_F32 (fp32 A/B, fp32 accum).
// ---------------------------------------------------------------------------

typedef float v2f __attribute__((ext_vector_type(2)));
typedef float v8f __attribute__((ext_vector_type(8)));

#define IMG   56
#define CH    128
#define WSZ   7
#define DISP  3
#define TOK   49     // 7*7 tokens per window
#define TOKP  64     // padded tokens
#define QKVC  384
#define HEADS 4
#define HD    32

// LDS layout (in floats). Strides padded so (stride mod 64 banks) == 4.
#define XS  132                       // x / attn row stride (128 cols)
#define QS  388                       // qkv row stride (384 cols)
#define PS  68                        // per-wave P slab row stride (64 cols)
#define OFF_POS 0                     // 169 floats (13x13), rounded to 172
#define OFF_XA  172                   // 64*132 = 8448 (x, later aliased: attn)
#define OFF_QKV (OFF_XA + TOKP*XS)    // 64*388 = 24832
#define OFF_P   (OFF_QKV + TOKP*QS)   // 8 waves * 16*68 = 8704
#define SMEM_FLOATS (OFF_P + 8*16*PS) // 42156
#define SMEM_BYTES  (SMEM_FLOATS * 4) // 168624 bytes < 320KB WGP LDS

// D = A(16x4) * B(4x16) + C, fp32. A: lane%16 = M, lane/16 selects K pair.
// B: lane%16 = N, lane/16 selects K pair. C/D: VGPR i -> M = i + 8*(lane/16),
// lane%16 = N (per ISA 7.12.2).
static __device__ inline v8f wmma4(v2f a, v2f b, v8f c) {
  return __builtin_amdgcn_wmma_f32_16x16x4_f32(
      /*neg_a=*/false, a, /*neg_b=*/false, b,
      /*c_mod=*/(short)0, c, /*reuse_a=*/false, /*reuse_b=*/false);
}

__global__ __launch_bounds__(256, 1)
void swin_window_attn_fused(const float* __restrict__ x,
                            const float* __restrict__ w_qkv,
                            const float* __restrict__ pos,
                            const float* __restrict__ w_out,
                            const float* __restrict__ b_out,
                            float* __restrict__ out) {
  extern __shared__ float smem[];
  float* lds_pos = smem + OFF_POS;
  float* lds_xa  = smem + OFF_XA;    // phase 0/1: x rows; phase 2/3: attn
  float* lds_qkv = smem + OFF_QKV;
  float* lds_p   = smem + OFF_P;

  const int tid  = threadIdx.x;
  const int lane = tid & 31;
  const int wv   = tid >> 5;       // wave id 0..7
  const int half = lane >> 4;      // K-pair / M+8 selector
  const int l16  = lane & 15;
  const int koff = half * 2;

  const int blk = blockIdx.x;      // b*64 + win
  const int b   = blk >> 6;
  const int win = blk & 63;
  const int wy  = win >> 3, wx = win & 7;

  // ---------------- Phase 0: stage pos_embedding + rolled/windowed x -------
  if (tid < 169) lds_pos[tid] = pos[tid];
  {
    const int r  = tid >> 2;          // token row 0..63
    const int c0 = (tid & 3) * 32;    // 32-float chunk
    float4* dst = (float4*)&lds_xa[r * XS + c0];
    if (r < TOK) {
      const int iy = r / WSZ, ix = r % WSZ;
      int hh = wy * WSZ + iy + DISP; if (hh >= IMG) hh -= IMG;
      int ww = wx * WSZ + ix + DISP; if (ww >= IMG) ww -= IMG;
      const float4* src =
          (const float4*)(x + (((size_t)b * IMG + hh) * IMG + ww) * CH + c0);
#pragma unroll
      for (int q = 0; q < 8; ++q) dst[q] = src[q];
    } else {
      float4 z = {0.f, 0.f, 0.f, 0.f};
#pragma unroll
      for (int q = 0; q < 8; ++q) dst[q] = z;   // zero pad rows
    }
  }
  __syncthreads();

  // ---------------- Phase 1: qkv = x @ w_qkv^T  (64x384, K=128) ------------
  __builtin_prefetch(&w_qkv[(wv * 48) * CH], 0, 2);
#pragma unroll 1
  for (int t3 = 0; t3 < 3; ++t3) {
    const int ct   = wv + t3 * 8;         // 24 column tiles of 16
    const int ncol = ct * 16 + l16;       // output channel 0..383
    v8f acc[4] = {{0,0,0,0,0,0,0,0}, {0,0,0,0,0,0,0,0},
                  {0,0,0,0,0,0,0,0}, {0,0,0,0,0,0,0,0}};
#pragma unroll 4
    for (int ks = 0; ks < 32; ++ks) {     // K = 128 in steps of 4
      // B[k][n] = w_qkv[n][k]
      v2f bf = *(const v2f*)&w_qkv[ncol * CH + ks * 4 + koff];
#pragma unroll
      for (int rt = 0; rt < 4; ++rt) {
        v2f af = *(const v2f*)&lds_xa[(rt * 16 + l16) * XS + ks * 4 + koff];
        acc[rt] = wmma4(af, bf, acc[rt]);
      }
    }
#pragma unroll
    for (int rt = 0; rt < 4; ++rt)
#pragma unroll
      for (int i = 0; i < 8; ++i)
        lds_qkv[(rt * 16 + i + half * 8) * QS + ct * 16 + l16] = acc[rt][i];
  }
  __syncthreads();

  // ---------------- Phase 2: attention per (head, row-tile) unit -----------
  const bool mUL = (wy == WSZ);   // same test as maskUL; unused
  const bool maskUL = (wy == 7);  // last window row: upper/lower shift mask
  const bool maskLR = (wx == 7);  // last window col: left/right shift mask
  (void)mUL;
  float* myP = lds_p + wv * 16 * PS;

#pragma unroll 1
  for (int rep = 0; rep < 2; ++rep) {
    const int u    = wv * 2 + rep;   // 16 units = 4 heads x 4 row tiles
    const int head = u >> 2;
    const int rt   = u & 3;
    const int qcol = head * HD;

    // S = q @ k^T   (16 rows x 64 cols, K = 32)
    v8f s[4];
#pragma unroll
    for (int ct = 0; ct < 4; ++ct) {
      v8f a = {0,0,0,0,0,0,0,0};
#pragma unroll
      for (int ks = 0; ks < 8; ++ks) {
        v2f af = *(const v2f*)&lds_qkv[(rt * 16 + l16) * QS + qcol + ks * 4 + koff];
        v2f bf = *(const v2f*)&lds_qkv[(ct * 16 + l16) * QS + CH + qcol + ks * 4 + koff];
        a = wmma4(af, bf, a);
      }
      s[ct] = a;
    }

    // scale + relative position bias + shift masks + pad masking
#pragma unroll
    for (int ct = 0; ct < 4; ++ct) {
      const int j  = ct * 16 + l16;
      const int jy = j / WSZ, jx = j % WSZ;
      const bool jv = (j < TOK);
#pragma unroll
      for (int i = 0; i < 8; ++i) {
        const int irow = rt * 16 + i + half * 8;
        float val;
        if (!jv) {
          val = -INFINITY;                       // padded keys
        } else if (irow >= TOK) {
          val = 0.0f;                            // padded queries: keep finite
        } else {
          const int iy = irow / WSZ, ix = irow % WSZ;
          val = s[ct][i] * 0.5f +                // SCALE = HEADS^-0.5 = 0.5
                lds_pos[(jy - iy + 6) * 13 + (jx - ix + 6)];
          if (maskUL && ((iy >= 4) != (jy >= 4))) val = -INFINITY;
          if (maskLR && ((ix >= 4) != (jx >= 4))) val = -INFINITY;
        }
        s[ct][i] = val;
      }
    }

    // softmax along j: row lives in 16 lanes of one half across 4 tiles
#pragma unroll
    for (int i = 0; i < 8; ++i) {
      float m = fmaxf(fmaxf(s[0][i], s[1][i]), fmaxf(s[2][i], s[3][i]));
      m = fmaxf(m, __shfl_xor(m, 1));
      m = fmaxf(m, __shfl_xor(m, 2));
      m = fmaxf(m, __shfl_xor(m, 4));
      m = fmaxf(m, __shfl_xor(m, 8));
      float e0 = __expf(s[0][i] - m), e1 = __expf(s[1][i] - m);
      float e2 = __expf(s[2][i] - m), e3 = __expf(s[3][i] - m);
      float sum = e0 + e1 + e2 + e3;
      sum += __shfl_xor(sum, 1);
      sum += __shfl_xor(sum, 2);
      sum += __shfl_xor(sum, 4);
      sum += __shfl_xor(sum, 8);
      const float inv = 1.0f / sum;
      s[0][i] = e0 * inv; s[1][i] = e1 * inv;
      s[2][i] = e2 * inv; s[3][i] = e3 * inv;
    }

    // reshape P (C-layout) -> A-layout through per-wave LDS slab
#pragma unroll
    for (int ct = 0; ct < 4; ++ct)
#pragma unroll
      for (int i = 0; i < 8; ++i)
        myP[(i + half * 8) * PS + ct * 16 + l16] = s[ct][i];

    // attn = P @ V   (16 x 32, K = 64)
    v8f o0 = {0,0,0,0,0,0,0,0}, o1 = {0,0,0,0,0,0,0,0};
#pragma unroll 4
    for (int ks = 0; ks < 16; ++ks) {
      v2f af = *(const v2f*)&myP[l16 * PS + ks * 4 + koff];
      const int j0 = ks * 4 + koff;
      v2f bf0, bf1;
      bf0.x = lds_qkv[ j0      * QS + 2 * CH + qcol + l16];
      bf0.y = lds_qkv[(j0 + 1) * QS + 2 * CH + qcol + l16];
      bf1.x = lds_qkv[ j0      * QS + 2 * CH + qcol + 16 + l16];
      bf1.y = lds_qkv[(j0 + 1) * QS + 2 * CH + qcol + 16 + l16];
      o0 = wmma4(af, bf0, o0);
      o1 = wmma4(af, bf1, o1);
    }
#pragma unroll
    for (int i = 0; i < 8; ++i) {
      lds_xa[(rt * 16 + i + half * 8) * XS + qcol +      l16] = o0[i];
      lds_xa[(rt * 16 + i + half * 8) * XS + qcol + 16 + l16] = o1[i];
    }
  }
  __syncthreads();

  // ---------------- Phase 3: out = attn @ w_out^T + b_out, inverse roll ----
  {
    const int ncol = wv * 16 + l16;      // output channel 0..127
    __builtin_prefetch(&w_out[ncol * CH], 0, 2);
    v8f acc[4] = {{0,0,0,0,0,0,0,0}, {0,0,0,0,0,0,0,0},
                  {0,0,0,0,0,0,0,0}, {0,0,0,0,0,0,0,0}};
#pragma unroll 4
    for (int ks = 0; ks < 32; ++ks) {
      v2f bf = *(const v2f*)&w_out[ncol * CH + ks * 4 + koff];
#pragma unroll
      for (int rt = 0; rt < 4; ++rt) {
        v2f af = *(const v2f*)&lds_xa[(rt * 16 + l16) * XS + ks * 4 + koff];
        acc[rt] = wmma4(af, bf, acc[rt]);
      }
    }
    const float bo = b_out[ncol];
#pragma unroll
    for (int rt = 0; rt < 4; ++rt) {
#pragma unroll
      for (int i = 0; i < 8; ++i) {
        const int tok = rt * 16 + i + half * 8;
        if (tok < TOK) {
          const int iy = tok / WSZ, ix = tok % WSZ;
          int hh = wy * WSZ + iy + DISP; if (hh >= IMG) hh -= IMG;
          int ww = wx * WSZ + ix + DISP; if (ww >= IMG) ww -= IMG;
          out[(((size_t)b * IMG + hh) * IMG + ww) * CH + ncol] = acc[rt][i] + bo;
        }
      }
    }
  }
}

extern "C" void kernel_launch(void* const* d_in, const int* in_sizes, int n_in,
                              void* d_out, int out_size, void* d_ws, size_t ws_size,
                              hipStream_t stream) {
  (void)in_sizes; (void)n_in; (void)out_size; (void)d_ws; (void)ws_size;
  const float* x     = (const float*)d_in[0];
  const float* w_qkv = (const float*)d_in[1];
  const float* pos   = (const float*)d_in[2];
  const float* w_out = (const float*)d_in[3];
  const float* b_out = (const float*)d_in[4];
  float* out = (float*)d_out;

  // gfx1250 WGP supports 320KB LDS; raise the dynamic-LDS cap (idempotent).
  (void)hipFuncSetAttribute((const void*)swin_window_attn_fused,
                            hipFuncAttributeMaxDynamicSharedMemorySize,
                            SMEM_BYTES);

  // 64 batches x 64 windows, 256 threads = 8 wave32 waves.
  swin_window_attn_fused<<<dim3(64 * 64), dim3(256), SMEM_BYTES, stream>>>(
      x, w_qkv, pos, w_out, b_out, out);
}